// SpKBGATModified_84859963834574
// MI455X (gfx1250) — compile-verified
//
#include <hip/hip_runtime.h>
#include <math.h>

// ---------------------------------------------------------------------------
// Problem constants (from the reference)
// ---------------------------------------------------------------------------
constexpr int N_      = 20000;
constexpr int IN_     = 128;
constexpr int NHID_   = 128;
constexpr int RELA_   = 128;
constexpr int NHEADS_ = 2;
constexpr int NREL_   = 500;
constexpr int E_      = 200000;
constexpr int E2_     = 25000;
constexpr int ET_     = E_ + E2_;   // 225000 total edges
constexpr int D2_     = 256;        // NHEADS*NHID

// Workspace layout (float offsets)
constexpr size_t XA_OFS   = 0;                      // [2][N][128]  -> reused as ya [N][256]
constexpr size_t XB_OFS   = XA_OFS + (size_t)2 * N_ * 128;   // [2][N][128] -> yb
constexpr size_t ACC_OFS  = XB_OFS + (size_t)2 * N_ * 128;   // [2][N][128] -> acc2 [N][256]
constexpr size_t LX_OFS   = ACC_OFS + (size_t)2 * N_ * 128;  // layer_x [N][256]
constexpr size_t RELP_OFS = LX_OFS + (size_t)N_ * D2_;       // [2][500][128] -> relp2 [500][256]
constexpr size_t RS_OFS   = RELP_OFS + (size_t)2 * NREL_ * 128; // rowsum [2][N] -> [N]

typedef __attribute__((ext_vector_type(2))) float v2f;
typedef __attribute__((ext_vector_type(8))) float v8f;

// ---------------------------------------------------------------------------
// C(MxNcol) = A(MxK, row-major, lda) @ W(Ncol x K, row-major, ldw)^T
// via V_WMMA_F32_16X16X4_F32. One 16x16 tile per wave, 4 waves / block.
// A-frag: lane(0-15)=row M, VGPR0/1 = K,K+1; upper half-lane K+2,K+3.
// B-frag: lane(0-15)=col N, same K striping (B[k][n] = W[n][k]).
// D: VGPR g -> row g (+8 for upper half), col = lane&15.
// ---------------------------------------------------------------------------
__global__ __launch_bounds__(128) void gemm_nt_f32wmma(
    const float* __restrict__ A, int lda,
    const float* __restrict__ W, int ldw,
    float* __restrict__ C, int ldc,
    int M, int Ncol, int K)
{
  const int ntn   = Ncol >> 4;
  const int mt    = ((M + 15) >> 4);
  const int tile  = blockIdx.x * 4 + (threadIdx.x >> 5);
  if (tile >= mt * ntn) return;            // whole-wave uniform exit (EXEC stays all-1)
  const int tm   = tile / ntn;
  const int tn   = tile % ntn;
  const int lane = threadIdx.x & 31;
  const int half = lane >> 4;
  const int l15  = lane & 15;

  int ar = tm * 16 + l15; if (ar >= M) ar = M - 1;      // clamp pad rows (stores guarded)
  const float* Ap = A + (size_t)ar * lda + half * 2;
  const float* Wp = W + (size_t)(tn * 16 + l15) * ldw + half * 2;

  v8f acc = {0.f, 0.f, 0.f, 0.f, 0.f, 0.f, 0.f, 0.f};
  for (int k = 0; k < K; k += 4) {
    v2f a = *(const v2f*)(Ap + k);
    v2f b = *(const v2f*)(Wp + k);
    acc = __builtin_amdgcn_wmma_f32_16x16x4_f32(false, a, false, b, (short)0, acc,
                                                false, false);
  }
  const int rbase = tm * 16 + half * 8;
  const int col   = tn * 16 + l15;
#pragma unroll
  for (int g = 0; g < 8; ++g) {
    int r = rbase + g;
    if (r < M) C[(size_t)r * ldc + col] = acc[g];
  }
}

// C(MxNcol) = A(MxK) @ B(K x Ncol, row-major, ldb)   (for rel_emb @ WR)
__global__ __launch_bounds__(128) void gemm_nn_f32wmma(
    const float* __restrict__ A, int lda,
    const float* __restrict__ B, int ldb,
    float* __restrict__ C, int ldc,
    int M, int Ncol, int K)
{
  const int ntn   = Ncol >> 4;
  const int mt    = ((M + 15) >> 4);
  const int tile  = blockIdx.x * 4 + (threadIdx.x >> 5);
  if (tile >= mt * ntn) return;
  const int tm   = tile / ntn;
  const int tn   = tile % ntn;
  const int lane = threadIdx.x & 31;
  const int half = lane >> 4;
  const int l15  = lane & 15;

  int ar = tm * 16 + l15; if (ar >= M) ar = M - 1;
  const float* Ap = A + (size_t)ar * lda + half * 2;
  const float* Bp = B + (size_t)(half * 2) * ldb + tn * 16 + l15;

  v8f acc = {0.f, 0.f, 0.f, 0.f, 0.f, 0.f, 0.f, 0.f};
  for (int k = 0; k < K; k += 4) {
    v2f a = *(const v2f*)(Ap + k);
    v2f b;
    b.x = Bp[(size_t)k * ldb];
    b.y = Bp[(size_t)k * ldb + ldb];
    acc = __builtin_amdgcn_wmma_f32_16x16x4_f32(false, a, false, b, (short)0, acc,
                                                false, false);
  }
  const int rbase = tm * 16 + half * 8;
  const int col   = tn * 16 + l15;
#pragma unroll
  for (int g = 0; g < 8; ++g) {
    int r = rbase + g;
    if (r < M) C[(size_t)r * ldc + col] = acc[g];
  }
}

// ---------------------------------------------------------------------------
// Fused edge pass, layer 1 (both heads). One wave per edge; 4 floats / lane.
// m = xa[src] + xb[dst] + relp[t0] (+ relp[t1]); ee = exp(-leakyrelu(m.w2));
// atomically accumulate ee*m into acc[h][src] and ee into rowsum[h][src].
// ---------------------------------------------------------------------------
__device__ __forceinline__ void decode_edge(int wid, const int* el, const int* et,
                                            const int* eln, const int* etn,
                                            int& src, int& dst, int& t0, int& t1) {
  if (wid < E_) {
    src = el[wid]; dst = el[E_ + wid]; t0 = et[wid]; t1 = -1;
  } else {
    int i = wid - E_;
    src = eln[i]; dst = eln[E2_ + i];
    t0 = etn[2 * i]; t1 = etn[2 * i + 1];
  }
}

__global__ __launch_bounds__(256) void edge_pass1(
    const float* __restrict__ xa, const float* __restrict__ xb,
    const float* __restrict__ relp, const float* __restrict__ w2h,
    const int* __restrict__ el, const int* __restrict__ et,
    const int* __restrict__ eln, const int* __restrict__ etn,
    float* __restrict__ acc, float* __restrict__ rowsum)
{
  const int wid  = (blockIdx.x * blockDim.x + threadIdx.x) >> 5;
  const int lane = threadIdx.x & 31;
  if (wid >= ET_) return;
  int src, dst, t0, t1;
  decode_edge(wid, el, et, eln, etn, src, dst, t0, t1);
  const int c = lane * 4;
#pragma unroll
  for (int h = 0; h < NHEADS_; ++h) {
    const float4 va = *(const float4*)(xa + ((size_t)h * N_ + src) * 128 + c);
    const float4 vb = *(const float4*)(xb + ((size_t)h * N_ + dst) * 128 + c);
    const float4 r0 = *(const float4*)(relp + ((size_t)h * NREL_ + t0) * 128 + c);
    float4 m;
    m.x = va.x + vb.x + r0.x;  m.y = va.y + vb.y + r0.y;
    m.z = va.z + vb.z + r0.z;  m.w = va.w + vb.w + r0.w;
    if (t1 >= 0) {
      const float4 r1 = *(const float4*)(relp + ((size_t)h * NREL_ + t1) * 128 + c);
      m.x += r1.x; m.y += r1.y; m.z += r1.z; m.w += r1.w;
    }
    const float4 w = *(const float4*)(w2h + h * 128 + c);
    float p = m.x * w.x + m.y * w.y + m.z * w.z + m.w * w.w;
#pragma unroll
    for (int off = 16; off > 0; off >>= 1) p += __shfl_xor(p, off, 32);
    const float z  = p > 0.f ? p : 0.2f * p;       // leaky_relu, alpha=0.2
    const float ee = expf(-z);
    float* ap = acc + ((size_t)h * N_ + src) * 128 + c;
    atomicAdd(ap + 0, ee * m.x);
    atomicAdd(ap + 1, ee * m.y);
    atomicAdd(ap + 2, ee * m.z);
    atomicAdd(ap + 3, ee * m.w);
    if (lane == 0) atomicAdd(rowsum + (size_t)h * N_ + src, ee);
  }
}

// Fused edge pass, layer 2 (256-wide). One wave per edge; 8 floats / lane.
__global__ __launch_bounds__(256) void edge_pass2(
    const float* __restrict__ ya, const float* __restrict__ yb,
    const float* __restrict__ relp2, const float* __restrict__ w2o,
    const int* __restrict__ el, const int* __restrict__ et,
    const int* __restrict__ eln, const int* __restrict__ etn,
    float* __restrict__ acc, float* __restrict__ rowsum)
{
  const int wid  = (blockIdx.x * blockDim.x + threadIdx.x) >> 5;
  const int lane = threadIdx.x & 31;
  if (wid >= ET_) return;
  int src, dst, t0, t1;
  decode_edge(wid, el, et, eln, etn, src, dst, t0, t1);
  const int c = lane * 8;

  float4 m0, m1;
  {
    const float4 a0 = *(const float4*)(ya + (size_t)src * D2_ + c);
    const float4 a1 = *(const float4*)(ya + (size_t)src * D2_ + c + 4);
    const float4 b0 = *(const float4*)(yb + (size_t)dst * D2_ + c);
    const float4 b1 = *(const float4*)(yb + (size_t)dst * D2_ + c + 4);
    const float4 r0 = *(const float4*)(relp2 + (size_t)t0 * D2_ + c);
    const float4 r1 = *(const float4*)(relp2 + (size_t)t0 * D2_ + c + 4);
    m0.x = a0.x + b0.x + r0.x; m0.y = a0.y + b0.y + r0.y;
    m0.z = a0.z + b0.z + r0.z; m0.w = a0.w + b0.w + r0.w;
    m1.x = a1.x + b1.x + r1.x; m1.y = a1.y + b1.y + r1.y;
    m1.z = a1.z + b1.z + r1.z; m1.w = a1.w + b1.w + r1.w;
  }
  if (t1 >= 0) {
    const float4 s0 = *(const float4*)(relp2 + (size_t)t1 * D2_ + c);
    const float4 s1 = *(const float4*)(relp2 + (size_t)t1 * D2_ + c + 4);
    m0.x += s0.x; m0.y += s0.y; m0.z += s0.z; m0.w += s0.w;
    m1.x += s1.x; m1.y += s1.y; m1.z += s1.z; m1.w += s1.w;
  }
  const float4 w0 = *(const float4*)(w2o + c);
  const float4 w1 = *(const float4*)(w2o + c + 4);
  float p = m0.x * w0.x + m0.y * w0.y + m0.z * w0.z + m0.w * w0.w
          + m1.x * w1.x + m1.y * w1.y + m1.z * w1.z + m1.w * w1.w;
#pragma unroll
  for (int off = 16; off > 0; off >>= 1) p += __shfl_xor(p, off, 32);
  const float z  = p > 0.f ? p : 0.2f * p;
  const float ee = expf(-z);
  float* ap = acc + (size_t)src * D2_ + c;
  atomicAdd(ap + 0, ee * m0.x); atomicAdd(ap + 1, ee * m0.y);
  atomicAdd(ap + 2, ee * m0.z); atomicAdd(ap + 3, ee * m0.w);
  atomicAdd(ap + 4, ee * m1.x); atomicAdd(ap + 5, ee * m1.y);
  atomicAdd(ap + 6, ee * m1.z); atomicAdd(ap + 7, ee * m1.w);
  if (lane == 0) atomicAdd(rowsum + src, ee);
}

// layer_x[n][h*128+j] = acc[h][n][j] / max(rowsum[h][n], 1e-12)
__global__ __launch_bounds__(256) void finalize1(
    const float* __restrict__ acc, const float* __restrict__ rs,
    float* __restrict__ lx)
{
  const size_t i = (size_t)blockIdx.x * blockDim.x + threadIdx.x;
  if (i >= (size_t)2 * N_ * 128) return;
  const int    h   = (int)(i / ((size_t)N_ * 128));
  const size_t rem = i - (size_t)h * N_ * 128;
  const int    n   = (int)(rem >> 7);
  const int    j   = (int)(rem & 127);
  float s = rs[(size_t)h * N_ + n];
  s = s < 1e-12f ? 1e-12f : s;
  lx[(size_t)n * D2_ + h * 128 + j] = acc[i] / s;
}

// out_entity = elu(elu(acc/rowsum))
__global__ __launch_bounds__(256) void finalize2(
    const float* __restrict__ acc, const float* __restrict__ rs,
    float* __restrict__ out)
{
  const size_t i = (size_t)blockIdx.x * blockDim.x + threadIdx.x;
  if (i >= (size_t)N_ * D2_) return;
  const int n = (int)(i >> 8);
  float s = rs[n];
  s = s < 1e-12f ? 1e-12f : s;
  const float hp = acc[i] / s;
  const float e1 = hp > 0.f ? hp : expm1f(hp);
  const float e2 = e1 > 0.f ? e1 : expm1f(e1);
  out[i] = e2;
}

// ---------------------------------------------------------------------------
extern "C" void kernel_launch(void* const* d_in, const int* in_sizes, int n_in,
                              void* d_out, int out_size, void* d_ws, size_t ws_size,
                              hipStream_t stream) {
  (void)in_sizes; (void)n_in; (void)out_size; (void)ws_size;
  const float* x        = (const float*)d_in[0];
  const float* rel_emb  = (const float*)d_in[1];
  const float* w1_heads = (const float*)d_in[2];   // [2][128][384]
  const float* w2_heads = (const float*)d_in[3];   // [2][128]
  const float* WR       = (const float*)d_in[4];   // [128][256]
  const float* w1_out   = (const float*)d_in[5];   // [256][768]
  const float* w2_out   = (const float*)d_in[6];   // [256]
  const int*   el       = (const int*)d_in[7];     // [2][E]
  const int*   et       = (const int*)d_in[8];     // [E]
  const int*   eln      = (const int*)d_in[9];     // [2][E2]
  const int*   etn      = (const int*)d_in[10];    // [E2][2]

  float* out     = (float*)d_out;
  float* out_rel = out + (size_t)N_ * D2_;         // out_relation [500][256]
  float* ws      = (float*)d_ws;

  float* xa   = ws + XA_OFS;    // [2][N][128]   (reused as ya [N][256])
  float* xb   = ws + XB_OFS;    // [2][N][128]   (reused as yb [N][256])
  float* acc  = ws + ACC_OFS;   // [2][N][128]   (reused as acc2 [N][256])
  float* lx   = ws + LX_OFS;    // [N][256]
  float* relp = ws + RELP_OFS;  // [2][500][128] (reused as relp2 [500][256])
  float* rs   = ws + RS_OFS;    // [2][N]        (reused as rowsum2 [N])

  const dim3 tpb128(128);
  auto gblocks = [](int tiles) { return (tiles + 3) / 4; };

  // ---- zero layer-1 accumulators
  hipMemsetAsync(acc, 0, (size_t)2 * N_ * 128 * sizeof(float), stream);
  hipMemsetAsync(rs,  0, (size_t)2 * N_ * sizeof(float), stream);

  // ---- layer-1 projections (WMMA fp32): xa/xb per head + relation projections
  for (int h = 0; h < NHEADS_; ++h) {
    const float* w1h = w1_heads + (size_t)h * NHID_ * 384;
    gemm_nt_f32wmma<<<gblocks((N_ / 16) * 8), tpb128, 0, stream>>>(
        x, IN_, w1h, 384, xa + (size_t)h * N_ * 128, 128, N_, 128, 128);
    gemm_nt_f32wmma<<<gblocks((N_ / 16) * 8), tpb128, 0, stream>>>(
        x, IN_, w1h + 128, 384, xb + (size_t)h * N_ * 128, 128, N_, 128, 128);
    gemm_nt_f32wmma<<<gblocks(32 * 8), tpb128, 0, stream>>>(
        rel_emb, RELA_, w1h + 256, 384, relp + (size_t)h * NREL_ * 128, 128,
        NREL_, 128, 128);
  }

  // ---- out_relation = rel_emb @ WR  (written straight into d_out tail)
  gemm_nn_f32wmma<<<gblocks(32 * 16), tpb128, 0, stream>>>(
      rel_emb, RELA_, WR, D2_, out_rel, D2_, NREL_, D2_, RELA_);

  // ---- layer-1 fused edge pass + finalize
  const int eblocks = (ET_ * 32 + 255) / 256;    // 28125
  edge_pass1<<<eblocks, 256, 0, stream>>>(xa, xb, relp, w2_heads,
                                          el, et, eln, etn, acc, rs);
  finalize1<<<(int)(((size_t)2 * N_ * 128 + 255) / 256), 256, 0, stream>>>(acc, rs, lx);

  // ---- layer-2 relation projection: relp2 = out_relation @ w1c_out^T
  gemm_nt_f32wmma<<<gblocks(32 * 16), tpb128, 0, stream>>>(
      out_rel, D2_, w1_out + 512, 768, relp, D2_, NREL_, D2_, D2_);

  // ---- layer-2 node projections: ya/yb = layer_x @ w1{a,b}_out^T
  float* ya = xa;
  float* yb = xb;
  gemm_nt_f32wmma<<<gblocks((N_ / 16) * 16), tpb128, 0, stream>>>(
      lx, D2_, w1_out, 768, ya, D2_, N_, D2_, D2_);
  gemm_nt_f32wmma<<<gblocks((N_ / 16) * 16), tpb128, 0, stream>>>(
      lx, D2_, w1_out + 256, 768, yb, D2_, N_, D2_, D2_);

  // ---- zero layer-2 accumulators, fused edge pass, finalize into d_out
  hipMemsetAsync(acc, 0, (size_t)N_ * D2_ * sizeof(float), stream);
  hipMemsetAsync(rs,  0, (size_t)N_ * sizeof(float), stream);
  edge_pass2<<<eblocks, 256, 0, stream>>>(ya, yb, relp, w2_out,
                                          el, et, eln, etn, acc, rs);
  finalize2<<<(int)(((size_t)N_ * D2_ + 255) / 256), 256, 0, stream>>>(acc, rs, out);
}